// BiLSTM_CRF_47012712022488
// MI455X (gfx1250) — compile-verified
//
#include <hip/hip_runtime.h>
#include <stdint.h>
#include <stddef.h>

// Batched Viterbi decode: B=256, T=2048, K=64, wave32 / gfx1250.
// One 64-thread block (2 waves) per batch element. Thread j owns next-state j
// and keeps its transitions row trans[j, 0..63] in 64 VGPRs.
// fv is double-buffered in LDS (broadcast reads), backpointers live in LDS
// (128KB -- only possible thanks to CDNA5's 320KB WGP LDS), and feats are
// staged LDS-resident through the Tensor Data Mover (double-buffered ring).

#define T_LEN   2048
#define K_TAGS  64
#define CH      32                 // timesteps per TDM chunk (8KB per buffer)
#define NCHUNK  (T_LEN / CH)
#define NEG_BIG (-3.0e38f)

typedef unsigned int v4u __attribute__((ext_vector_type(4)));
typedef int          v8i __attribute__((ext_vector_type(8)));
typedef int          v4i __attribute__((ext_vector_type(4)));

#if defined(__gfx1250__) && __has_builtin(__builtin_amdgcn_tensor_load_to_lds)
#define USE_TDM 1
#else
#define USE_TDM 0
#endif

#if USE_TDM
// Build a 1-D TDM descriptor (D#) and issue tensor_load_to_lds (6-arg form,
// clang-23 / therock headers).
// Group0: count=1 | lds_addr | global_addr[31:0] | global_addr[56:32] + type=2
// Group1: data_size=4B, tensor_dim0=n, tensor_dim1=1, tile_dim0=n (1-D tile),
//         tile_dim1=tile_dim2=0, stride0=n. Groups 2/3 (and trailing group)
//         unused -> zeros.
__device__ __forceinline__ void tdm_load_chunk(uint32_t lds_off,
                                               const float* gsrc, int n) {
  uint64_t ga = (uint64_t)(uintptr_t)gsrc;
  v4u g0;
  g0.x = 1u;                                              // count = 1 (valid)
  g0.y = lds_off;                                         // LDS byte address
  g0.z = (uint32_t)ga;                                    // global addr lo
  g0.w = (uint32_t)((ga >> 32) & 0x01FFFFFFu) | (2u << 30); // addr hi | type=2
  v8i g1;
  g1[0] = (int)(2u << 16);                                // data_size = 4B
  g1[1] = (int)(((uint32_t)n & 0xFFFFu) << 16);           // tensor_dim0[15:0]
  g1[2] = (int)(((uint32_t)n >> 16) | (1u << 16));        // dim0 hi | tensor_dim1=1
  g1[3] = (int)(((uint32_t)n & 0xFFFFu) << 16);           // tile_dim0 = n
  g1[4] = 0;                                              // tile_dim1=0, tile_dim2=0
  g1[5] = n;                                              // tensor_dim0_stride lo
  g1[6] = 0;
  g1[7] = 0;
  v4i g2 = {0, 0, 0, 0};
  v4i g3 = {0, 0, 0, 0};
  v8i g4 = {0, 0, 0, 0, 0, 0, 0, 0};
  __builtin_amdgcn_tensor_load_to_lds(g0, g1, g2, g3, g4, 0);
}
#endif

__global__ __launch_bounds__(K_TAGS) void viterbi_kernel(
    const float* __restrict__ feats, const float* __restrict__ trans,
    const int* __restrict__ start_p, const int* __restrict__ stop_p,
    float* __restrict__ out_score, float* __restrict__ out_path) {
  extern __shared__ __align__(16) char smem[];
  float*   fvbuf = (float*)smem;                              // 2*64 floats
  float*   term  = fvbuf + 2 * K_TAGS;                        // 64 floats
  float*   ring  = (float*)(smem + 1024);                     // 2*CH*64 floats
  uint8_t* bp    = (uint8_t*)(smem + 1024 + 2 * CH * K_TAGS * 4); // 128KB

  const int j = threadIdx.x;          // next-state owned by this thread
  const int b = blockIdx.x;           // batch element
  const int start_tag = *start_p;
  const int stop_tag  = *stop_p;

  // Per-thread transitions row: tr[p] = trans[j*K + p] (stays in VGPRs).
  float tr[K_TAGS];
  {
    const float4* t4 = (const float4*)(trans + (size_t)j * K_TAGS);
#pragma unroll
    for (int q = 0; q < K_TAGS / 4; ++q) {
      float4 v = t4[q];
      tr[4 * q + 0] = v.x; tr[4 * q + 1] = v.y;
      tr[4 * q + 2] = v.z; tr[4 * q + 3] = v.w;
    }
  }

  fvbuf[j] = (j == start_tag) ? 0.0f : -10000.0f;   // init vector (t = -1)

  const float* fb = feats + (size_t)b * T_LEN * K_TAGS;

#if USE_TDM
  if (j < 32) {  // TDM ignores EXEC: issue from wave 0 only
    tdm_load_chunk((uint32_t)(uintptr_t)ring, fb, CH * K_TAGS);
  }
#endif
  __syncthreads();

  for (int c = 0; c < NCHUNK; ++c) {
#if USE_TDM
    if (j < 32) {
      if (c + 1 < NCHUNK) {
        tdm_load_chunk((uint32_t)(uintptr_t)(ring + ((c + 1) & 1) * CH * K_TAGS),
                       fb + (size_t)(c + 1) * CH * K_TAGS, CH * K_TAGS);
        __builtin_amdgcn_s_wait_tensorcnt(1);  // chunk c landed (in-order)
      } else {
        __builtin_amdgcn_s_wait_tensorcnt(0);
      }
    }
    __syncthreads();
    const float* fchunk = ring + (c & 1) * CH * K_TAGS;
#else
    const float* fchunk = fb + (size_t)c * CH * K_TAGS;
    __builtin_prefetch(fb + (size_t)(c + 1) * CH * K_TAGS + j * 4, 0, 0);
#endif

    for (int tt = 0; tt < CH; ++tt) {
      const int t = c * CH + tt;
      const float4* fv4 = (const float4*)(fvbuf + (t & 1) * K_TAGS);
      const float fval = fchunk[tt * K_TAGS + j];   // feats[b,t,j]

      // 64-wide max-plus argmax, 4 independent chains (mod-4) for ILP.
      float b0 = NEG_BIG, b1 = NEG_BIG, b2 = NEG_BIG, b3 = NEG_BIG;
      int   i0 = 0, i1 = 1, i2 = 2, i3 = 3;
#pragma unroll
      for (int q = 0; q < K_TAGS / 4; ++q) {
        float4 v = fv4[q];   // broadcast LDS read of fv[4q..4q+3]
        float s0 = v.x + tr[4 * q + 0]; if (s0 > b0) { b0 = s0; i0 = 4 * q + 0; }
        float s1 = v.y + tr[4 * q + 1]; if (s1 > b1) { b1 = s1; i1 = 4 * q + 1; }
        float s2 = v.z + tr[4 * q + 2]; if (s2 > b2) { b2 = s2; i2 = 4 * q + 2; }
        float s3 = v.w + tr[4 * q + 3]; if (s3 > b3) { b3 = s3; i3 = 4 * q + 3; }
      }
      // Merge chains; on exact ties prefer the smaller prev index
      // (matches jnp.argmax "first occurrence").
      if (b1 > b0 || (b1 == b0 && i1 < i0)) { b0 = b1; i0 = i1; }
      if (b3 > b2 || (b3 == b2 && i3 < i2)) { b2 = b3; i2 = i3; }
      if (b2 > b0 || (b2 == b0 && i2 < i0)) { b0 = b2; i0 = i2; }

      bp[t * K_TAGS + j] = (uint8_t)i0;               // LDS-resident backptr
      fvbuf[((t + 1) & 1) * K_TAGS + j] = b0 + fval;  // fv_new = max + feat
      __syncthreads();
    }
  }

  // Terminal: fv_last + trans[stop_tag, :]
  term[j] = fvbuf[(T_LEN & 1) * K_TAGS + j] + trans[(size_t)stop_tag * K_TAGS + j];
  __syncthreads();

  if (j == 0) {
    float bs = term[0];
    int   bi = 0;
    for (int p = 1; p < K_TAGS; ++p) {
      float s = term[p];
      if (s > bs) { bs = s; bi = p; }   // strict >: first index on ties
    }
    out_score[b] = bs;

    // Backtrace entirely from LDS; path[t] = tag, then tag = bp[t][tag].
    int tag = bi;
    float* op = out_path + (size_t)b * T_LEN;
    for (int t = T_LEN - 1; t >= 0; --t) {
      op[t] = (float)tag;
      tag = bp[t * K_TAGS + tag];
    }
  }
}

extern "C" void kernel_launch(void* const* d_in, const int* in_sizes, int n_in,
                              void* d_out, int out_size, void* d_ws, size_t ws_size,
                              hipStream_t stream) {
  const float* feats = (const float*)d_in[0];
  const float* trans = (const float*)d_in[1];
  const int* startp  = (const int*)d_in[2];
  const int* stopp   = (const int*)d_in[3];
  float* out = (float*)d_out;

  const int B = in_sizes[0] / (T_LEN * K_TAGS);   // 256

  const size_t shmem = 1024                         // fv double buffer + term
                     + (size_t)2 * CH * K_TAGS * 4  // TDM feat ring (16KB)
                     + (size_t)T_LEN * K_TAGS;      // backpointers (128KB)

  // Opt in to >64KB dynamic LDS (CDNA5: 320KB/WGP). Host-side, capture-safe.
  (void)hipFuncSetAttribute((const void*)viterbi_kernel,
                            hipFuncAttributeMaxDynamicSharedMemorySize,
                            (int)shmem);

  viterbi_kernel<<<B, K_TAGS, shmem, stream>>>(feats, trans, startp, stopp,
                                               out, out + B);
}